// CountSketch2_9414568313530
// MI455X (gfx1250) — compile-verified
//
#include <hip/hip_runtime.h>

// Tensor-sketch (compact bilinear pooling), fully fused per batch row:
//   count-sketch -> FFT8192 -> Hermitian split + complex product -> IFFT8192
// One workgroup (256 threads = 8 wave32) per row; data lives in LDS.
//
// tcFFT structure: 12 of the 13 radix-2 stages run as three radix-16
// "super-stages" on the f32 WMMA pipe (V_WMMA_F32_16X16X4_F32):
//   out[g*16S + m*S + o] = sum_t G[m][t] * (W_{16S}^(sgn*o*br4(t)) * cur[g*16S + t*S + o])
// with G[m][t] = W16^(sgn*m*br4(t)), for strides S = 1, 16, 256.
// Only the final len-8192 radix-2 stage runs on VALU. All twiddles are exact
// shift-indexed loads from a shared 4096-entry table (no transcendentals in
// the steady state, no divisions).

#define N_BATCH 4096
#define D_IN    4096
#define M_SK    8192
#define LOG2M   13
#define THREADS 256
#define NWAVES  (THREADS / 32)

typedef __attribute__((ext_vector_type(2))) float v2f;
typedef __attribute__((ext_vector_type(8))) float v8f;

__device__ __forceinline__ int bitrev13(int x) {
  return (int)(__brev((unsigned)x) >> 19);
}
__device__ __forceinline__ int bitrev4(int x) {
  return (int)(__brev((unsigned)x) >> 28);
}

// One radix-16 DIT super-stage (replaces 4 radix-2 stages) on the WMMA pipe.
// lS = log2(element stride S). Columns c = (group, offset) pairs; each wave
// owns whole 16-column tiles, so reads/writes are wave-private between barriers.
__device__ __forceinline__ void radix16_wmma(float* re, float* im,
                                             const float2* tw, float sgn,
                                             const int lS) {
  const int lane = threadIdx.x & 31;
  const int wave = threadIdx.x >> 5;
  const int m    = lane & 15;   // A row (freq digit) == B/D column lane
  const int kh   = lane >> 4;   // K half-select: VGPR.x -> K = 4c + 2*kh
  const int S    = 1 << lS;
  const int twsh = 8 - lS;      // o*br4(t) * (256/S) indexes the 4096 table

  for (int tile = wave; tile < 32; tile += NWAVES) {
    const int c     = tile * 16 + m;              // this lane's column
    const int o     = c & (S - 1);                // offset within 16S-group
    const int cbase = ((c & ~(S - 1)) << 4) | o;  // element t lives at cbase + t*S

    // B operands: twiddle-premultiplied data  B'[t][c] = W_{16S}^(sgn*o*br4(t)) * z[...]
    v2f br_[4], bi_[4];
    #pragma unroll
    for (int cc = 0; cc < 4; ++cc) {
      const int t0 = 4 * cc + 2 * kh;
      #pragma unroll
      for (int u = 0; u < 2; ++u) {
        const int t = t0 + u;
        const int a = cbase + (t << lS);
        const float  xr = re[a], xi = im[a];
        const float2 w  = tw[(o * bitrev4(t)) << twsh];
        const float  ws = sgn * w.y;
        const float  pr = xr * w.x - xi * ws;
        const float  pi = xr * ws + xi * w.x;
        if (u == 0) { br_[cc].x = pr; bi_[cc].x = pi; }
        else        { br_[cc].y = pr; bi_[cc].y = pi; }
      }
    }

    v8f dr = {};  // real accumulator
    v8f di = {};  // imag accumulator
    #pragma unroll
    for (int cc = 0; cc < 4; ++cc) {
      const int k0 = 4 * cc + 2 * kh;
      // A operands: G row m, K columns k0, k0+1; exact mod-16 phase -> p*256.
      const int p0 = (m * bitrev4(k0))     & 15;
      const int p1 = (m * bitrev4(k0 + 1)) & 15;
      const float2 w0 = tw[p0 << 8];
      const float2 w1 = tw[p1 << 8];
      v2f gr;  gr.x  =  w0.x;        gr.y  =  w1.x;
      v2f gi;  gi.x  =  sgn * w0.y;  gi.y  =  sgn * w1.y;
      v2f gin; gin.x = -gi.x;        gin.y = -gi.y;
      // (Gr + i*Gi)(Br + i*Bi):  Dr = Gr*Br - Gi*Bi ; Di = Gi*Br + Gr*Bi
      dr = __builtin_amdgcn_wmma_f32_16x16x4_f32(false, gr,  false, br_[cc], (short)0, dr, false, false);
      dr = __builtin_amdgcn_wmma_f32_16x16x4_f32(false, gin, false, bi_[cc], (short)0, dr, false, false);
      di = __builtin_amdgcn_wmma_f32_16x16x4_f32(false, gi,  false, br_[cc], (short)0, di, false, false);
      di = __builtin_amdgcn_wmma_f32_16x16x4_f32(false, gr,  false, bi_[cc], (short)0, di, false, false);
    }

    // D: lane column c, rows r + 8*kh at stride S
    #pragma unroll
    for (int r = 0; r < 8; ++r) {
      const int idx = cbase + ((r + 8 * kh) << lS);
      re[idx] = dr[r];
      im[idx] = di[r];
    }
  }
}

__device__ void fft8192(float* re, float* im, const float2* tw, float sgn) {
  const int tid = threadIdx.x;

  // Bit-reversal permutation (each pair swapped by exactly one thread).
  for (int i = tid; i < M_SK; i += THREADS) {
    const int j = bitrev13(i);
    if (j > i) {
      float tr = re[i]; re[i] = re[j]; re[j] = tr;
      float ti = im[i]; im[i] = im[j]; im[j] = ti;
    }
  }
  __syncthreads();

  radix16_wmma(re, im, tw, sgn, 0);   // stages len = 2 .. 16
  __syncthreads();
  radix16_wmma(re, im, tw, sgn, 4);   // stages len = 32 .. 256
  __syncthreads();
  radix16_wmma(re, im, tw, sgn, 8);   // stages len = 512 .. 4096
  __syncthreads();

  // Final radix-2 stage, len = 8192 (16 butterflies per thread on VALU).
  for (int t = tid; t < (M_SK >> 1); t += THREADS) {
    const float2 w = tw[t];
    const float c = w.x;
    const float s = sgn * w.y;
    const float ur = re[t],                  ui = im[t];
    const float vr = re[t + (M_SK >> 1)],    vi = im[t + (M_SK >> 1)];
    const float tr = vr * c - vi * s;
    const float ti = vr * s + vi * c;
    re[t]               = ur + tr;  im[t]               = ui + ti;
    re[t + (M_SK >> 1)] = ur - tr;  im[t + (M_SK >> 1)] = ui - ti;
  }
  __syncthreads();
}

__global__ void __launch_bounds__(THREADS)
tensor_sketch_fft_kernel(const float* __restrict__ x,     const float* __restrict__ y,
                         const float* __restrict__ sign1, const int* __restrict__ indx1,
                         const float* __restrict__ sign2, const int* __restrict__ indx2,
                         float* __restrict__ out) {
  __shared__ float  s_re[M_SK];       // re = count-sketch(x), then spectrum/result
  __shared__ float  s_im[M_SK];       // im = count-sketch(y)  (pack-2-reals trick)
  __shared__ float2 s_tw[M_SK / 2];   // tw[j] = (cos, sin)(2*pi*j / 8192)

  const int tid = threadIdx.x;
  const int row = blockIdx.x;
  const float* xrow = x + (size_t)row * D_IN;
  const float* yrow = y + (size_t)row * D_IN;

  // Prefetch input rows (global_prefetch_b8), one lane per 256B line.
  for (int i = tid * 64; i < D_IN; i += THREADS * 64) {
    __builtin_prefetch(xrow + i, 0, 0);
    __builtin_prefetch(yrow + i, 0, 0);
  }

  // Build twiddle table (16 sincos per thread, once) + zero the sketch arrays.
  for (int j = tid; j < M_SK / 2; j += THREADS) {
    float s, c;
    __sincosf(7.6699039e-4f * (float)j, &s, &c);   // 2*pi/8192 * j
    s_tw[j] = make_float2(c, s);
  }
  for (int i = tid; i < M_SK; i += THREADS) { s_re[i] = 0.0f; s_im[i] = 0.0f; }
  __syncthreads();

  // Count sketches via LDS atomics (ds_add_f32): z = CS(x) + i*CS(y).
  for (int i = tid; i < D_IN; i += THREADS) {
    atomicAdd(&s_re[indx1[i]], xrow[i] * sign1[i]);
    atomicAdd(&s_im[indx2[i]], yrow[i] * sign2[i]);
  }
  __syncthreads();

  fft8192(s_re, s_im, s_tw, -1.0f);   // forward FFT of packed signal

  // Split packed spectrum (X = even part, Y = odd part), multiply, write back
  // Hermitian product so the inverse transform of a real signal is exact.
  for (int k = tid; k <= M_SK / 2; k += THREADS) {
    const int k2 = (M_SK - k) & (M_SK - 1);
    const float ar = s_re[k],  ai = s_im[k];
    const float br = s_re[k2], bi = s_im[k2];
    const float Xr = 0.5f * (ar + br), Xi = 0.5f * (ai - bi);
    const float Yr = 0.5f * (ai + bi), Yi = 0.5f * (br - ar);
    const float Pr = Xr * Yr - Xi * Yi;
    const float Pi = Xr * Yi + Xi * Yr;
    s_re[k]  = Pr;  s_im[k]  = Pi;
    s_re[k2] = Pr;  s_im[k2] = -Pi;
  }
  __syncthreads();

  fft8192(s_re, s_im, s_tw, +1.0f);   // inverse FFT (unscaled)

  const float scale = 1.0f / (float)M_SK;
  float* orow = out + (size_t)row * M_SK;
  for (int i = tid; i < M_SK; i += THREADS) orow[i] = s_re[i] * scale;
}

extern "C" void kernel_launch(void* const* d_in, const int* in_sizes, int n_in,
                              void* d_out, int out_size, void* d_ws, size_t ws_size,
                              hipStream_t stream) {
  (void)in_sizes; (void)n_in; (void)d_ws; (void)ws_size; (void)out_size;
  const float* x     = (const float*)d_in[0];
  const float* y     = (const float*)d_in[1];
  const float* sign1 = (const float*)d_in[2];
  const int*   indx1 = (const int*)d_in[3];
  const float* sign2 = (const float*)d_in[4];
  const int*   indx2 = (const int*)d_in[5];
  float* out = (float*)d_out;

  dim3 grid(N_BATCH), block(THREADS);
  hipLaunchKernelGGL(tensor_sketch_fft_kernel, grid, block, 0, stream,
                     x, y, sign1, indx1, sign2, indx2, out);
}